// LRFDescriptor_56521769616144
// MI455X (gfx1250) — compile-verified
//
#include <hip/hip_runtime.h>
#include <hip/hip_bf16.h>

typedef __attribute__((ext_vector_type(2))) float v2f;
typedef __attribute__((ext_vector_type(8))) float v8f;

#define BATCH 8
#define NPTS  4096
#define KNN   16

// out layout (floats): lrfs [8,4096,3,3] | nbh_proj [8,4096,16,3] | nbh_idx [8,4096,16] | lengths [8]
#define LRFS_OFF 0
#define PROJ_OFF (BATCH * NPTS * 9)                  // 294912
#define IDX_OFF  (PROJ_OFF + BATCH * NPTS * KNN * 3) // 1867776
#define LEN_OFF  (IDX_OFF + BATCH * NPTS * KNN)      // 2392064

// ---------------------------------------------------------------------------
// Kernel 1: brute-force kNN per cloud using V_WMMA_F32_16X16X4_F32 tiles.
//   Distance trick using the full K=4: A row m = (x, y, z, |p_m|^2),
//   B col n = (-2qx, -2qy, -2qz, 1)  =>  acc[m,n] = |p_m|^2 - 2 p_m.q_n,
//   d2 = acc + |q_n|^2  (per-lane scalar) -- no per-tile norm loads needed.
// Grid: 8 batches * 32 chunks = 256 blocks of 256 threads (8 waves).
// Each wave owns 16 queries (query q handled by lane pair {q%16, q%16+16}).
// ---------------------------------------------------------------------------
__global__ __launch_bounds__(256) void knn_kernel(const float* __restrict__ points,
                                                  int* __restrict__ knn_out) {
    __shared__ float4 pts[NPTS]; // x,y,z,|p|^2  -> 64KB LDS

    const int b     = blockIdx.x >> 5;
    const int chunk = blockIdx.x & 31;
    const int tid   = threadIdx.x;

    // cooperative stage of the whole cloud into LDS
    for (int i = tid; i < NPTS; i += 256) {
        const float* p = points + ((size_t)b * NPTS + i) * 3;
        float x = p[0], y = p[1], z = p[2];
        pts[i] = make_float4(x, y, z, x * x + y * y + z * z);
    }
    __syncthreads();

    const int  lane = tid & 31;
    const int  wv   = tid >> 5;
    const int  lh   = lane & 15;
    const bool hi   = lane >= 16;

    const int q = chunk * 128 + wv * 16 + lh;   // this lane-pair's query
    float4 qp = pts[q];

    // B operand (queries as columns): lanes 0-15 -> (-2x,-2y) = K0,K1 ;
    //                                 lanes 16-31 -> (-2z, 1) = K2,K3
    v2f Breg;
    Breg.x = hi ? (-2.0f * qp.z) : (-2.0f * qp.x);
    Breg.y = hi ? 1.0f           : (-2.0f * qp.y);
    const float sqq = qp.w;

    float bd[KNN];
    int   bi[KNN];
#pragma unroll
    for (int i = 0; i < KNN; ++i) { bd[i] = 3.4e38f; bi[i] = 0; }

    // 2x unrolled tile loop: two back-to-back WMMAs so the insertion-filter
    // VALU work co-executes with the matrix pipe.
    for (int mb = 0; mb < NPTS; mb += 32) {
#pragma unroll
        for (int t = 0; t < 2; ++t) {
            const int tb = mb + t * 16;
            float4 mp = pts[tb + lh];
            // A operand (DB points as rows): lanes 0-15 -> (x,y), lanes 16-31 -> (z,|p|^2)
            v2f Areg;
            Areg.x = hi ? mp.z : mp.x;
            Areg.y = hi ? mp.w : mp.y;

            v8f acc = {};
            acc = __builtin_amdgcn_wmma_f32_16x16x4_f32(
                /*neg_a=*/false, Areg, /*neg_b=*/false, Breg,
                /*c_mod=*/(short)0, acc, /*reuse_a=*/false, /*reuse_b=*/false);

#pragma unroll
            for (int r = 0; r < 8; ++r) {
                int   m  = tb + r + (hi ? 8 : 0);
                float d2 = acc[r] + sqq;
                if (d2 < bd[KNN - 1]) {
                    float dd = d2; int ii = m;
#pragma unroll
                    for (int j = 0; j < KNN; ++j) {   // register insertion sort
                        bool  sw = dd < bd[j];
                        float td = bd[j]; int ti = bi[j];
                        bd[j] = sw ? dd : td;  bi[j] = sw ? ii : ti;
                        dd    = sw ? td : dd;  ii    = sw ? ti : ii;
                    }
                }
            }
        }
    }

    // merge the two half-lists (lane l has top-16 of rows r%16<8, lane l+16 the rest):
    // bitonic lowest-16: L[i] = min(A[i], B[15-i]), then bitonic merge network.
    float md[KNN];
    int   mi[KNN];
#pragma unroll
    for (int i = 0; i < KNN; ++i) {
        float pd = __shfl_xor(bd[15 - i], 16, 32);
        int   pi = __shfl_xor(bi[15 - i], 16, 32);
        bool  take = pd < bd[i];
        md[i] = take ? pd : bd[i];
        mi[i] = take ? pi : bi[i];
    }
#pragma unroll
    for (int k = 8; k >= 1; k >>= 1) {
#pragma unroll
        for (int i = 0; i < KNN; ++i) {
            if ((i & k) == 0) {
                int   j  = i | k;
                bool  sw = md[j] < md[i];
                float ti = md[i]; int ii = mi[i];
                float tj = md[j]; int ij = mi[j];
                md[i] = sw ? tj : ti;  mi[i] = sw ? ij : ii;
                md[j] = sw ? ti : tj;  mi[j] = sw ? ii : ij;
            }
        }
    }

    if (!hi) {
        size_t base = ((size_t)b * NPTS + q) * KNN;
#pragma unroll
        for (int i = 0; i < KNN; ++i) knn_out[base + i] = mi[i];
    }
}

// ---------------------------------------------------------------------------
// Kernel 2: per-point covariance -> closed-form 3x3 eigh -> disambiguated LRF
// -> neighbor projections. One thread per point (32768 threads).
// ---------------------------------------------------------------------------
__device__ __forceinline__ void eig_vec(float a00, float a01, float a02,
                                        float a11, float a12, float a22,
                                        float lam, float& vx, float& vy, float& vz) {
    float m00 = a00 - lam, m11 = a11 - lam, m22 = a22 - lam;
    // cross products of rows of (A - lam I); largest-norm one spans the null space
    float c0x = a01 * a12 - a02 * m11, c0y = a02 * a01 - m00 * a12, c0z = m00 * m11 - a01 * a01;
    float c1x = a01 * m22 - a02 * a12, c1y = a02 * a02 - m00 * m22, c1z = m00 * a12 - a01 * a02;
    float c2x = m11 * m22 - a12 * a12, c2y = a12 * a02 - a01 * m22, c2z = a01 * a12 - m11 * a02;
    float n0 = c0x * c0x + c0y * c0y + c0z * c0z;
    float n1 = c1x * c1x + c1y * c1y + c1z * c1z;
    float n2 = c2x * c2x + c2y * c2y + c2z * c2z;
    float x = c0x, y = c0y, z = c0z, n = n0;
    if (n1 > n) { x = c1x; y = c1y; z = c1z; n = n1; }
    if (n2 > n) { x = c2x; y = c2y; z = c2z; n = n2; }
    float inv = rsqrtf(fmaxf(n, 1e-30f));
    vx = x * inv; vy = y * inv; vz = z * inv;
}

__global__ __launch_bounds__(256) void lrf_kernel(const float* __restrict__ points,
                                                  const int* __restrict__ knn,
                                                  float* __restrict__ out) {
    int gid = blockIdx.x * blockDim.x + threadIdx.x;
    if (gid < BATCH) out[LEN_OFF + gid] = (float)NPTS;
    if (gid >= BATCH * NPTS) return;
    int b = gid / NPTS;

    const float* pp = points + (size_t)gid * 3;
    float px = pp[0], py = pp[1], pz = pp[2];

    float nx[KNN], ny[KNN], nz[KNN];
    int   idxs[KNN];
    float mx = 0.f, my = 0.f, mz = 0.f;
#pragma unroll
    for (int k = 0; k < KNN; ++k) {
        int m = knn[(size_t)gid * KNN + k];
        idxs[k] = m;
        const float* np = points + ((size_t)b * NPTS + m) * 3;
        nx[k] = np[0]; ny[k] = np[1]; nz[k] = np[2];
        mx += nx[k]; my += ny[k]; mz += nz[k];
    }
    const float invK = 1.0f / (float)KNN;
    mx *= invK; my *= invK; mz *= invK;

    float a00 = 0.f, a01 = 0.f, a02 = 0.f, a11 = 0.f, a12 = 0.f, a22 = 0.f;
#pragma unroll
    for (int k = 0; k < KNN; ++k) {
        float dx = nx[k] - mx, dy = ny[k] - my, dz = nz[k] - mz;
        a00 += dx * dx; a01 += dx * dy; a02 += dx * dz;
        a11 += dy * dy; a12 += dy * dz; a22 += dz * dz;
    }
    a00 *= invK; a01 *= invK; a02 *= invK; a11 *= invK; a12 *= invK; a22 *= invK;

    // analytic symmetric 3x3 eigenvalues (trigonometric form)
    float p1 = a01 * a01 + a02 * a02 + a12 * a12;
    float qt = (a00 + a11 + a22) * (1.0f / 3.0f);
    float d0 = a00 - qt, d1 = a11 - qt, d2 = a22 - qt;
    float p2 = d0 * d0 + d1 * d1 + d2 * d2 + 2.0f * p1;
    float pr = sqrtf(fmaxf(p2 * (1.0f / 6.0f), 1e-30f));
    float ip = 1.0f / pr;
    float b00 = d0 * ip, b01 = a01 * ip, b02 = a02 * ip;
    float b11 = d1 * ip, b12 = a12 * ip, b22 = d2 * ip;
    float detB = b00 * (b11 * b22 - b12 * b12)
               - b01 * (b01 * b22 - b12 * b02)
               + b02 * (b01 * b12 - b11 * b02);
    float r = fminf(fmaxf(0.5f * detB, -1.0f), 1.0f);
    float phi = acosf(r) * (1.0f / 3.0f);
    float emax = qt + 2.0f * pr * cosf(phi);
    float emin = qt + 2.0f * pr * cosf(phi + 2.09439510239f); // + 2*pi/3

    float zx, zy, zz_, xx_, xy_, xz_;
    eig_vec(a00, a01, a02, a11, a12, a22, emin, zx, zy, zz_);  // normal: smallest eig
    eig_vec(a00, a01, a02, a11, a12, a22, emax, xx_, xy_, xz_); // tangent: largest eig

    // neighbor offsets df = nbh - p ; disambiguate both directions
    float dfx[KNN], dfy[KNN], dfz[KNN];
    int nposz = 0, nposx = 0;
#pragma unroll
    for (int k = 0; k < KNN; ++k) {
        dfx[k] = nx[k] - px; dfy[k] = ny[k] - py; dfz[k] = nz[k] - pz;
        float prz = dfx[k] * zx + dfy[k] * zy + dfz[k] * zz_;
        float prx = dfx[k] * xx_ + dfy[k] * xy_ + dfz[k] * xz_;
        nposz += (prz > 0.0f) ? 1 : 0;
        nposx += (prx > 0.0f) ? 1 : 0;
    }
    float sz = (nposz < KNN / 2) ? -1.0f : 1.0f;
    float sx = (nposx < KNN / 2) ? -1.0f : 1.0f;
    zx *= sz; zy *= sz; zz_ *= sz;
    xx_ *= sx; xy_ *= sx; xz_ *= sx;

    // y = z cross x
    float yx = zy * xz_ - zz_ * xy_;
    float yy = zz_ * xx_ - zx * xz_;
    float yz = zx * xy_ - zy * xx_;

    // lrfs[b,n,d,e] with columns (z,x,y)
    float* lr = out + LRFS_OFF + (size_t)gid * 9;
    lr[0] = zx;  lr[1] = xx_; lr[2] = yx;
    lr[3] = zy;  lr[4] = xy_; lr[5] = yy;
    lr[6] = zz_; lr[7] = xz_; lr[8] = yz;

    float* pj = out + PROJ_OFF + (size_t)gid * (KNN * 3);
    float* id = out + IDX_OFF + (size_t)gid * KNN;
#pragma unroll
    for (int k = 0; k < KNN; ++k) {
        pj[k * 3 + 0] = dfx[k] * zx + dfy[k] * zy + dfz[k] * zz_;
        pj[k * 3 + 1] = dfx[k] * xx_ + dfy[k] * xy_ + dfz[k] * xz_;
        pj[k * 3 + 2] = dfx[k] * yx + dfy[k] * yy + dfz[k] * yz;
        id[k] = (float)idxs[k];
    }
}

extern "C" void kernel_launch(void* const* d_in, const int* in_sizes, int n_in,
                              void* d_out, int out_size, void* d_ws, size_t ws_size,
                              hipStream_t stream) {
    (void)in_sizes; (void)n_in; (void)out_size; (void)ws_size;
    const float* points = (const float*)d_in[0];
    int*   knn_ws = (int*)d_ws;          // [B*N*K] neighbor indices (local per cloud)
    float* out    = (float*)d_out;

    knn_kernel<<<BATCH * (NPTS / 128), 256, 0, stream>>>(points, knn_ws);
    lrf_kernel<<<(BATCH * NPTS + 255) / 256, 256, 0, stream>>>(points, knn_ws, out);
}